// GATLayer_60902636257847
// MI455X (gfx1250) — compile-verified
//
#include <hip/hip_runtime.h>
#include <hip/hip_bf16.h>

#define N_NODES 100000
#define N_EDGES 1600000
#define IN_FEAT 128
#define HEADS 4
#define OUT_FEAT 16
#define HF 64               // HEADS * OUT_FEAT
#define LRELU_ALPHA 0.2f
#define ROW_BLOCKS ((N_NODES + 15) / 16)   // 6250 (N divisible by 16)

typedef __attribute__((ext_vector_type(2))) float v2f;
typedef __attribute__((ext_vector_type(8))) float v8f;

// ---------------------------------------------------------------------------
// monotone float <-> sortable uint encoding for atomic float-max
// ---------------------------------------------------------------------------
__device__ __forceinline__ unsigned enc_f(float f) {
    unsigned u = __float_as_uint(f);
    return (u & 0x80000000u) ? ~u : (u | 0x80000000u);
}
__device__ __forceinline__ float dec_f(unsigned u) {
    u = (u & 0x80000000u) ? (u & 0x7FFFFFFFu) : ~u;
    return __uint_as_float(u);
}
#define ENC_NEG_INF 0x007FFFFFu   // enc(-inf)

// ---------------------------------------------------------------------------
// 0) init: zero accumulator (d_out) and denom, seed emax with enc(-inf)
// ---------------------------------------------------------------------------
__global__ void gat_init(float* __restrict__ out, float* __restrict__ denom,
                         unsigned* __restrict__ emax_enc) {
    int i = blockIdx.x * blockDim.x + threadIdx.x;
    if (i < N_NODES * HF) out[i] = 0.0f;
    if (i < N_NODES * HEADS) { denom[i] = 0.0f; emax_enc[i] = ENC_NEG_INF; }
}

// ---------------------------------------------------------------------------
// 1) h = x @ W via V_WMMA_F32_16X16X4_F32. One wave -> one 16x64 tile of h.
//    W (128x64 f32 = 32KB) staged in LDS per workgroup.
//    A-frag (16x4 f32): lane half selects K pair {0,1}/{2,3}; M = lane&15.
//    B-frag (4x16 f32): VGPR j holds K = j + 2*half; N = lane&15.
// ---------------------------------------------------------------------------
__global__ __launch_bounds__(256) void gat_gemm(const float* __restrict__ x,
                                                const float* __restrict__ W,
                                                float* __restrict__ h) {
    __shared__ float Ws[IN_FEAT * HF];          // 32 KB of the 320 KB WGP LDS
    for (int i = threadIdx.x; i < (IN_FEAT * HF) / 4; i += 256)
        ((float4*)Ws)[i] = ((const float4*)W)[i];
    __syncthreads();

    const int wave = threadIdx.x >> 5;
    const int lane = threadIdx.x & 31;
    const int half = lane >> 4;                 // 0: lanes 0-15, 1: lanes 16-31
    const int l16  = lane & 15;
    const int rb   = blockIdx.x * 8 + wave;     // 16-row block index
    if (rb >= ROW_BLOCKS) return;

    const int row = rb * 16 + l16;
    const float* __restrict__ xrow = x + (size_t)row * IN_FEAT;

    v8f acc[4] = {};                            // 4 column tiles of 16 (one per head)
    for (int k = 0; k < IN_FEAT; k += 4) {
        const float2 av = *(const float2*)(xrow + k + 2 * half);
        v2f A = {av.x, av.y};
#pragma unroll
        for (int t = 0; t < 4; ++t) {
            v2f B = {Ws[(k + 2 * half) * HF + t * 16 + l16],
                     Ws[(k + 2 * half + 1) * HF + t * 16 + l16]};
            acc[t] = __builtin_amdgcn_wmma_f32_16x16x4_f32(
                false, A, false, B, (short)0, acc[t], false, false);
        }
    }

    // C/D layout: VGPR v -> row (v + 8*half), col = l16 within tile t
#pragma unroll
    for (int t = 0; t < 4; ++t)
#pragma unroll
        for (int v = 0; v < 8; ++v)
            h[(size_t)(rb * 16 + v + 8 * half) * HF + t * 16 + l16] = acc[t][v];
}

// ---------------------------------------------------------------------------
// 2) per-(node,head) attention logits: dot(h[n, hd*16 : hd*16+16], a_{src,dst}[hd])
// ---------------------------------------------------------------------------
__global__ void gat_att(const float* __restrict__ h, const float* __restrict__ a_src,
                        const float* __restrict__ a_dst, float* __restrict__ att_s,
                        float* __restrict__ att_d) {
    int i = blockIdx.x * blockDim.x + threadIdx.x;     // node*HEADS + head
    if (i >= N_NODES * HEADS) return;
    const int hd = i & (HEADS - 1);
    const float* __restrict__ hp = h + (size_t)(i >> 2) * HF + hd * OUT_FEAT;
    const float* __restrict__ as = a_src + hd * OUT_FEAT;
    const float* __restrict__ ad = a_dst + hd * OUT_FEAT;
    float s = 0.0f, d = 0.0f;
#pragma unroll
    for (int f = 0; f < OUT_FEAT; ++f) {
        const float v = hp[f];
        s += v * as[f];
        d += v * ad[f];
    }
    att_s[i] = s;
    att_d[i] = d;
}

// ---------------------------------------------------------------------------
// 3) edge pass 1: e = leakyrelu(att_s[src] + att_d[dst]); stash e; segment max
// ---------------------------------------------------------------------------
__global__ void gat_edge_max(const int* __restrict__ ei, const float* __restrict__ att_s,
                             const float* __restrict__ att_d, float* __restrict__ e_store,
                             unsigned* __restrict__ emax_enc) {
    int i = blockIdx.x * blockDim.x + threadIdx.x;     // edge*HEADS + head
    if (i >= N_EDGES * HEADS) return;
    const int e = i >> 2, hd = i & (HEADS - 1);
    const int src = ei[e];
    const int dst = ei[N_EDGES + e];
    float v = att_s[src * HEADS + hd] + att_d[dst * HEADS + hd];
    v = (v > 0.0f) ? v : (LRELU_ALPHA * v);
    e_store[i] = v;
    atomicMax(&emax_enc[dst * HEADS + hd], enc_f(v));
}

// ---------------------------------------------------------------------------
// 4) decode segment max in place; nodes with no in-edges -> 0 (per reference)
// ---------------------------------------------------------------------------
__global__ void gat_emax_fix(unsigned* __restrict__ emax_enc) {
    int i = blockIdx.x * blockDim.x + threadIdx.x;
    if (i >= N_NODES * HEADS) return;
    const float v = dec_f(emax_enc[i]);
    ((float*)emax_enc)[i] = __builtin_isfinite(v) ? v : 0.0f;
}

// ---------------------------------------------------------------------------
// 5) edge pass 2: ex = exp(e - emax[dst]); segment sum into denom
// ---------------------------------------------------------------------------
__global__ void gat_edge_exp(const int* __restrict__ ei, const float* __restrict__ emax,
                             float* __restrict__ ex, float* __restrict__ denom) {
    int i = blockIdx.x * blockDim.x + threadIdx.x;     // edge*HEADS + head
    if (i >= N_EDGES * HEADS) return;
    const int e = i >> 2, hd = i & (HEADS - 1);
    const int dst = ei[N_EDGES + e];
    const float v = expf(ex[i] - emax[dst * HEADS + hd]);
    ex[i] = v;
    atomicAdd(&denom[dst * HEADS + hd], v);
}

// ---------------------------------------------------------------------------
// 6) scatter: out[dst, hd*16+f] += h[src, hd*16+f] * ex/(denom+eps)
//    one thread per (edge, head, f): f is the fastest dim -> coalesced h reads
//    and atomics; dst working set lives in the 192MB L2.
// ---------------------------------------------------------------------------
__global__ void gat_scatter(const int* __restrict__ ei, const float* __restrict__ ex,
                            const float* __restrict__ denom, const float* __restrict__ h,
                            float* __restrict__ out) {
    int i = blockIdx.x * blockDim.x + threadIdx.x;     // < E*HEADS*OUT_FEAT = 102.4M
    if (i >= N_EDGES * HEADS * OUT_FEAT) return;
    const int f  = i & (OUT_FEAT - 1);
    const int eh = i >> 4;                              // edge*HEADS + head
    const int e  = eh >> 2, hd = eh & (HEADS - 1);
    const int src = ei[e];
    const int dst = ei[N_EDGES + e];
    const float alpha = ex[eh] / (denom[dst * HEADS + hd] + 1e-16f);
    const float msg = h[(size_t)src * HF + hd * OUT_FEAT + f] * alpha;
    atomicAdd(&out[(size_t)dst * HF + hd * OUT_FEAT + f], msg);
}

// ---------------------------------------------------------------------------
// 7) ELU in place on d_out
// ---------------------------------------------------------------------------
__global__ void gat_elu(float* __restrict__ out) {
    int i = blockIdx.x * blockDim.x + threadIdx.x;
    if (i >= N_NODES * HF) return;
    const float v = out[i];
    out[i] = (v > 0.0f) ? v : (expf(v) - 1.0f);
}

// ---------------------------------------------------------------------------
extern "C" void kernel_launch(void* const* d_in, const int* in_sizes, int n_in,
                              void* d_out, int out_size, void* d_ws, size_t ws_size,
                              hipStream_t stream) {
    const float* x     = (const float*)d_in[0];
    const int*   ei    = (const int*)d_in[1];     // edge_index [2, E]
    const float* W     = (const float*)d_in[2];
    const float* a_src = (const float*)d_in[3];
    const float* a_dst = (const float*)d_in[4];
    float* out = (float*)d_out;

    char* ws = (char*)d_ws;
    float* h = (float*)ws;              ws += sizeof(float) * (size_t)N_NODES * HF;
    float* att_s = (float*)ws;          ws += sizeof(float) * (size_t)N_NODES * HEADS;
    float* att_d = (float*)ws;          ws += sizeof(float) * (size_t)N_NODES * HEADS;
    unsigned* emax_enc = (unsigned*)ws; ws += sizeof(unsigned) * (size_t)N_NODES * HEADS;
    float* denom = (float*)ws;          ws += sizeof(float) * (size_t)N_NODES * HEADS;
    float* ex = (float*)ws;             ws += sizeof(float) * (size_t)N_EDGES * HEADS;

    const int TB = 256;
    gat_init<<<(N_NODES * HF + TB - 1) / TB, TB, 0, stream>>>(out, denom, emax_enc);
    gat_gemm<<<(ROW_BLOCKS + 7) / 8, TB, 0, stream>>>(x, W, h);
    gat_att<<<(N_NODES * HEADS + TB - 1) / TB, TB, 0, stream>>>(h, a_src, a_dst, att_s, att_d);
    gat_edge_max<<<(N_EDGES * HEADS + TB - 1) / TB, TB, 0, stream>>>(ei, att_s, att_d, ex, emax_enc);
    gat_emax_fix<<<(N_NODES * HEADS + TB - 1) / TB, TB, 0, stream>>>(emax_enc);
    gat_edge_exp<<<(N_EDGES * HEADS + TB - 1) / TB, TB, 0, stream>>>(
        ei, (const float*)emax_enc, ex, denom);
    gat_scatter<<<(N_EDGES * HEADS * OUT_FEAT + TB - 1) / TB, TB, 0, stream>>>(
        ei, ex, denom, h, out);
    gat_elu<<<(N_NODES * HF + TB - 1) / TB, TB, 0, stream>>>(out);
}